// GTATIntegrated_42588895707827
// MI455X (gfx1250) — compile-verified
//
#include <hip/hip_runtime.h>

// GTAT fused implementation for gfx1250 (MI455X).
// Big GEMMs use V_WMMA_F32_16X16X4_F32 (fp32 matrix core, matches fp32 reference).
// Fusions: Hp-GEMM eliminated (only Hp@wh needed -> vector u = Wf@wh);
//          h_new@Wo folded as alpha@(T_out@Wo) = alpha@M.

typedef float v2f __attribute__((ext_vector_type(2)));
typedef float v8f __attribute__((ext_vector_type(8)));

#define Dx 256
#define AS_STRIDE 260   // padded LDS stride for the 16x256 A tile

// ---------------------------------------------------------------------------
// WMMA core: C(16x256) += A(16x256, LDS, stride AS_STRIDE) @ B(256x256, global)
// Wave w covers cols [32w, 32w+32) as two 16-wide N tiles.
// Fragment layouts per CDNA5 ISA 7.12.2:
//   A 16x4:  lanes 0-15 -> M=lane, K = {v + 0}; lanes 16-31 -> M=lane-16, K = {v + 2}
//   B 4x16:  vgpr v holds rows K=v (lanes 0-15) and K=v+2 (lanes 16-31), N = lane%16
//   C 16x16: vgpr r holds M=r (lanes 0-15) and M=r+8 (lanes 16-31), N = lane%16
// ---------------------------------------------------------------------------
__device__ __forceinline__ void wmma_tile_16x256(const float* __restrict__ As,
                                                 const float* __restrict__ Bm,
                                                 int wave, int lo, int hi,
                                                 v8f& acc0, v8f& acc1)
{
    const int col0 = wave * 32 + lo;
    const int col1 = col0 + 16;
#pragma unroll 4
    for (int k = 0; k < Dx; k += 4) {
        const int ka = k + 2 * hi;
        v2f a;
        a.x = As[lo * AS_STRIDE + ka];
        a.y = As[lo * AS_STRIDE + ka + 1];
        v2f b0, b1;
        b0.x = Bm[ka * Dx + col0];
        b0.y = Bm[(ka + 1) * Dx + col0];
        b1.x = Bm[ka * Dx + col1];
        b1.y = Bm[(ka + 1) * Dx + col1];
        acc0 = __builtin_amdgcn_wmma_f32_16x16x4_f32(false, a, false, b0,
                                                     (short)0, acc0, false, false);
        acc1 = __builtin_amdgcn_wmma_f32_16x16x4_f32(false, a, false, b1,
                                                     (short)0, acc1, false, false);
    }
}

// ---------------------------------------------------------------------------
// Big GEMM: out = A @ B + bias, optional row LayerNorm.  16 rows per block.
// ---------------------------------------------------------------------------
template <bool DO_LN>
__global__ __launch_bounds__(256)
void bigmm_kernel(const float* __restrict__ A, const float* __restrict__ Bm,
                  const float* __restrict__ bias,
                  const float* __restrict__ ln_g, const float* __restrict__ ln_b,
                  float* __restrict__ out)
{
    __shared__ float As[16 * AS_STRIDE];
    __shared__ float red[256];
    __shared__ float red2[256];
    __shared__ float rowm[16];
    __shared__ float rowi[16];

    const int tid = threadIdx.x;
    const long rowbase = (long)blockIdx.x * 16;

#pragma unroll
    for (int i = 0; i < 16; ++i)
        As[i * AS_STRIDE + tid] = A[(rowbase + i) * Dx + tid];
    __syncthreads();

    const int wave = tid >> 5;
    const int lane = tid & 31;
    const int lo = lane & 15;
    const int hi = lane >> 4;
    const int col0 = wave * 32 + lo;
    const int col1 = col0 + 16;

    v8f acc0 = {0.f, 0.f, 0.f, 0.f, 0.f, 0.f, 0.f, 0.f};
    v8f acc1 = acc0;
    wmma_tile_16x256(As, Bm, wave, lo, hi, acc0, acc1);

    if (!DO_LN) {
#pragma unroll
        for (int r = 0; r < 8; ++r) {
            const int m = r + 8 * hi;
            out[(rowbase + m) * Dx + col0] = acc0[r] + bias[col0];
            out[(rowbase + m) * Dx + col1] = acc1[r] + bias[col1];
        }
        return;
    }

    __syncthreads();               // everyone done reading As
#pragma unroll
    for (int r = 0; r < 8; ++r) {  // stage pre-LN values (reuse As)
        const int m = r + 8 * hi;
        As[m * AS_STRIDE + col0] = acc0[r] + bias[col0];
        As[m * AS_STRIDE + col1] = acc1[r] + bias[col1];
    }
    __syncthreads();

    const int row = tid >> 4, seg = tid & 15;
    float s = 0.f, q = 0.f;
#pragma unroll
    for (int j = 0; j < 16; ++j) {
        const float v = As[row * AS_STRIDE + seg * 16 + j];
        s += v; q += v * v;
    }
    red[tid] = s; red2[tid] = q;
    __syncthreads();
    if (tid < 16) {
        float ss = 0.f, qq = 0.f;
#pragma unroll
        for (int j = 0; j < 16; ++j) { ss += red[tid * 16 + j]; qq += red2[tid * 16 + j]; }
        const float mean = ss * (1.0f / Dx);
        const float var = qq * (1.0f / Dx) - mean * mean;
        rowm[tid] = mean;
        rowi[tid] = rsqrtf(var + 1e-5f);
    }
    __syncthreads();
    const float mean = rowm[row], rinv = rowi[row];
#pragma unroll
    for (int j = 0; j < 16; ++j) {
        const int n = seg * 16 + j;
        const float v = As[row * AS_STRIDE + n];
        out[(rowbase + row) * Dx + n] = (v - mean) * rinv * ln_g[n] + ln_b[n];
    }
}

// ---------------------------------------------------------------------------
// Fused layer: per row  s = h.u + c ; alpha = softmax(lrelu(s + tvec + ba));
//              h = LN(alpha @ M + bo + h).   In place on h, 16 rows / block.
// ---------------------------------------------------------------------------
__global__ __launch_bounds__(256)
void layer_kernel(float* __restrict__ h, const float* __restrict__ M,
                  const float* __restrict__ u, const float* __restrict__ cps,
                  const float* __restrict__ tvec, const float* __restrict__ ba,
                  const float* __restrict__ bo,
                  const float* __restrict__ lng, const float* __restrict__ lnb)
{
    __shared__ float Hs[16 * Dx];
    __shared__ float As[16 * AS_STRIDE];
    __shared__ float red[256];
    __shared__ float red2[256];
    __shared__ float rowa[16];
    __shared__ float rowb[16];

    const int tid = threadIdx.x;
    const long rowbase = (long)blockIdx.x * 16;

#pragma unroll
    for (int i = 0; i < 16; ++i)
        Hs[i * Dx + tid] = h[(rowbase + i) * Dx + tid];
    __syncthreads();

    const int row = tid >> 4, seg = tid & 15;

    // ---- s = dot(h_row, u) + c   (16 threads per row) ----
    float ps = 0.f;
#pragma unroll
    for (int j = 0; j < 16; ++j) {
        const int n = seg * 16 + j;
        ps += Hs[row * Dx + n] * u[n];
    }
    red[tid] = ps;
    __syncthreads();
    if (tid < 16) {
        float ss = cps[0];
#pragma unroll
        for (int j = 0; j < 16; ++j) ss += red[tid * 16 + j];
        rowa[tid] = ss;                      // s per row
    }
    __syncthreads();

    // ---- e = leaky_relu(s + tvec + ba); softmax over 256 -> alpha in As ----
    const float srow = rowa[row];
    const float bav = ba[0];
    float lmax = -3.4e38f;
#pragma unroll
    for (int j = 0; j < 16; ++j) {
        const int n = seg * 16 + j;
        float v = srow + tvec[n] + bav;
        v = v > 0.f ? v : 0.01f * v;
        As[row * AS_STRIDE + n] = v;
        lmax = fmaxf(lmax, v);
    }
    red[tid] = lmax;
    __syncthreads();
    if (tid < 16) {
        float mx = red[tid * 16];
#pragma unroll
        for (int j = 1; j < 16; ++j) mx = fmaxf(mx, red[tid * 16 + j]);
        rowb[tid] = mx;
    }
    __syncthreads();
    const float mx = rowb[row];
    float lsum = 0.f;
#pragma unroll
    for (int j = 0; j < 16; ++j) {
        const int n = seg * 16 + j;
        const float p = __expf(As[row * AS_STRIDE + n] - mx);
        As[row * AS_STRIDE + n] = p;
        lsum += p;
    }
    red[tid] = lsum;
    __syncthreads();
    if (tid < 16) {
        float ss = 0.f;
#pragma unroll
        for (int j = 0; j < 16; ++j) ss += red[tid * 16 + j];
        rowa[tid] = 1.0f / ss;
    }
    __syncthreads();
    const float inv = rowa[row];
#pragma unroll
    for (int j = 0; j < 16; ++j)
        As[row * AS_STRIDE + seg * 16 + j] *= inv;
    __syncthreads();

    // ---- alpha @ M via WMMA ----
    const int wave = tid >> 5;
    const int lane = tid & 31;
    const int lo = lane & 15;
    const int hi = lane >> 4;
    const int col0 = wave * 32 + lo;
    const int col1 = col0 + 16;

    v8f acc0 = {0.f, 0.f, 0.f, 0.f, 0.f, 0.f, 0.f, 0.f};
    v8f acc1 = acc0;
    wmma_tile_16x256(As, M, wave, lo, hi, acc0, acc1);

    __syncthreads();               // alpha no longer needed; reuse As as stage
#pragma unroll
    for (int r = 0; r < 8; ++r) {
        const int m = r + 8 * hi;
        As[m * AS_STRIDE + col0] = acc0[r] + bo[col0] + Hs[m * Dx + col0];
        As[m * AS_STRIDE + col1] = acc1[r] + bo[col1] + Hs[m * Dx + col1];
    }
    __syncthreads();

    // ---- LayerNorm, write back in place ----
    float s = 0.f, q = 0.f;
#pragma unroll
    for (int j = 0; j < 16; ++j) {
        const float v = As[row * AS_STRIDE + seg * 16 + j];
        s += v; q += v * v;
    }
    red[tid] = s; red2[tid] = q;
    __syncthreads();
    if (tid < 16) {
        float ss = 0.f, qq = 0.f;
#pragma unroll
        for (int j = 0; j < 16; ++j) { ss += red[tid * 16 + j]; qq += red2[tid * 16 + j]; }
        const float mean = ss * (1.0f / Dx);
        const float var = qq * (1.0f / Dx) - mean * mean;
        rowa[tid] = mean;
        rowb[tid] = rsqrtf(var + 1e-5f);
    }
    __syncthreads();
    const float mean = rowa[row], rinv = rowb[row];
#pragma unroll
    for (int j = 0; j < 16; ++j) {
        const int n = seg * 16 + j;
        const float v = As[row * AS_STRIDE + n];
        h[(rowbase + row) * Dx + n] = (v - mean) * rinv * lng[n] + lnb[n];
    }
}

// ---------------------------------------------------------------------------
// Tiny precompute kernels
// ---------------------------------------------------------------------------
__global__ void prep_t_kernel(const float* __restrict__ gdv,
                              const float* __restrict__ g1W, const float* __restrict__ g1b,
                              const float* __restrict__ g2W, const float* __restrict__ g2b,
                              const float* __restrict__ grW, const float* __restrict__ grb,
                              float* __restrict__ t)
{
    const int r = threadIdx.x;       // 256 rows
    float sum = 0.f;
    for (int k = 0; k < 73; ++k) sum += gdv[r * 73 + k];
    const float inv = 1.0f / (sum + 1e-6f);
    float h1[64];
    for (int j = 0; j < 64; ++j) {
        float acc = g1b[j];
        for (int k = 0; k < 73; ++k) acc += gdv[r * 73 + k] * inv * g1W[k * 64 + j];
        h1[j] = fmaxf(acc, 0.f);
    }
    for (int j = 0; j < 64; ++j) {
        float acc = g2b[j] + grb[j];
        for (int k = 0; k < 64; ++k) acc += h1[k] * g2W[k * 64 + j];
        for (int k = 0; k < 73; ++k) acc += gdv[r * 73 + k] * inv * grW[k * 64 + j];
        t[r * 64 + j] = acc;
    }
}

__global__ void tp_kernel(const float* __restrict__ t, const float* __restrict__ Wt_l,
                          const float* __restrict__ bt_l, float* __restrict__ Tp)
{
    __shared__ float ts[64];
    const int b = blockIdx.x, j = threadIdx.x;
    if (j < 64) ts[j] = t[b * 64 + j];
    __syncthreads();
    float acc = bt_l[j];
    for (int k = 0; k < 64; ++k) acc += ts[k] * Wt_l[k * Dx + j];
    Tp[b * Dx + j] = acc;
}

__global__ void pvec_kernel(const float* __restrict__ Tp, const float* __restrict__ wa_topo_l,
                            float* __restrict__ pi, float* __restrict__ pj)
{
    const int r = threadIdx.x;
    const float* wi = wa_topo_l;
    const float* wj = wa_topo_l + Dx;
    float a = 0.f, b = 0.f;
    for (int k = 0; k < Dx; ++k) {
        const float v = Tp[r * Dx + k];
        a += v * wi[k]; b += v * wj[k];
    }
    pi[r] = a; pj[r] = b;
}

__global__ void beta_kernel(const float* __restrict__ pi, const float* __restrict__ pj,
                            const float* __restrict__ ba, const int* __restrict__ adj,
                            float* __restrict__ beta)
{
    __shared__ float red[256];
    const int r = blockIdx.x, j = threadIdx.x;
    float e;
    if (adj[r * Dx + j] == 0) e = -1e9f;
    else {
        const float v = pi[r] + pj[j] + ba[0];
        e = v > 0.f ? v : 0.01f * v;
    }
    red[j] = e;
    __syncthreads();
    for (int s2 = 128; s2 > 0; s2 >>= 1) {
        if (j < s2) red[j] = fmaxf(red[j], red[j + s2]);
        __syncthreads();
    }
    const float mx = red[0];
    __syncthreads();
    const float p = __expf(e - mx);
    red[j] = p;
    __syncthreads();
    for (int s2 = 128; s2 > 0; s2 >>= 1) {
        if (j < s2) red[j] += red[j + s2];
        __syncthreads();
    }
    beta[r * Dx + j] = p / red[0];
}

__global__ void matmul256_kernel(const float* __restrict__ A, const float* __restrict__ B,
                                 float* __restrict__ C)
{
    __shared__ float arow[256];
    const int r = blockIdx.x, j = threadIdx.x;
    arow[j] = A[r * Dx + j];
    __syncthreads();
    float acc = 0.f;
    for (int k = 0; k < Dx; ++k) acc += arow[k] * B[k * Dx + j];
    C[r * Dx + j] = acc;
}

__global__ void vec_kernel(const float* __restrict__ Tout, const float* __restrict__ wa_feat_l,
                           const float* __restrict__ Wf_l, const float* __restrict__ bf_l,
                           float* __restrict__ tvec, float* __restrict__ u,
                           float* __restrict__ cps)
{
    const int d = threadIdx.x;
    const float* wh = wa_feat_l;
    const float* wt = wa_feat_l + Dx;
    float a = 0.f, b = 0.f;
    for (int j = 0; j < Dx; ++j) {
        a += Tout[d * Dx + j] * wt[j];
        b += Wf_l[d * Dx + j] * wh[j];
    }
    tvec[d] = a; u[d] = b;
    if (d == 0) {
        float c = 0.f;
        for (int j = 0; j < Dx; ++j) c += bf_l[j] * wh[j];
        cps[0] = c;
    }
}

// ---------------------------------------------------------------------------
extern "C" void kernel_launch(void* const* d_in, const int* in_sizes, int n_in,
                              void* d_out, int out_size, void* d_ws, size_t ws_size,
                              hipStream_t stream)
{
    (void)in_sizes; (void)n_in; (void)out_size; (void)ws_size;

    const float* x       = (const float*)d_in[0];
    const int*   adj     = (const int*)  d_in[1];
    const float* gdv     = (const float*)d_in[2];
    const float* W_in    = (const float*)d_in[3];
    const float* b_in    = (const float*)d_in[4];
    const float* W_out   = (const float*)d_in[5];
    const float* b_out   = (const float*)d_in[6];
    const float* g1W     = (const float*)d_in[7];
    const float* g1b     = (const float*)d_in[8];
    const float* g2W     = (const float*)d_in[9];
    const float* g2b     = (const float*)d_in[10];
    const float* grW     = (const float*)d_in[11];
    const float* grb     = (const float*)d_in[12];
    const float* ln_g    = (const float*)d_in[13];
    const float* ln_b    = (const float*)d_in[14];
    const float* Wf      = (const float*)d_in[15];
    const float* bf      = (const float*)d_in[16];
    const float* Wt      = (const float*)d_in[17];
    const float* bt      = (const float*)d_in[18];
    const float* wa_feat = (const float*)d_in[19];
    const float* ba_feat = (const float*)d_in[20];
    const float* wa_topo = (const float*)d_in[21];
    const float* ba_topo = (const float*)d_in[22];
    const float* Wo      = (const float*)d_in[23];
    const float* bo      = (const float*)d_in[24];
    const float* lng     = (const float*)d_in[25];
    const float* lnb     = (const float*)d_in[26];

    // h lives in d_out (32768*256 floats); all big kernels update rows in place.
    float* h = (float*)d_out;

    // Small workspace buffers (~1.1 MB total)
    float* ws   = (float*)d_ws;
    float* t    = ws;             // 256*64
    float* Tp   = t + 256 * 64;   // 256*256
    float* pi   = Tp + 256 * 256; // 256
    float* pj   = pi + 256;       // 256
    float* beta = pj + 256;       // 256*256
    float* Tout = beta + 256 * 256; // 256*256
    float* M    = Tout + 256 * 256; // 256*256
    float* tvec = M + 256 * 256;  // 256
    float* u    = tvec + 256;     // 256
    float* cps  = u + 256;        // 16

    const int NROWS = 8 * 4096;          // 32768
    const int NBLK = NROWS / 16;         // 2048

    prep_t_kernel<<<1, 256, 0, stream>>>(gdv, g1W, g1b, g2W, g2b, grW, grb, t);
    bigmm_kernel<false><<<NBLK, 256, 0, stream>>>(x, W_in, b_in, nullptr, nullptr, h);

    for (int l = 0; l < 2; ++l) {
        tp_kernel<<<256, 256, 0, stream>>>(t, Wt + l * 64 * 256, bt + l * 256, Tp);
        pvec_kernel<<<1, 256, 0, stream>>>(Tp, wa_topo + l * 512, pi, pj);
        beta_kernel<<<256, 256, 0, stream>>>(pi, pj, ba_topo + l, adj, beta);
        matmul256_kernel<<<256, 256, 0, stream>>>(beta, Tp, Tout);          // T_out
        matmul256_kernel<<<256, 256, 0, stream>>>(Tout, Wo + l * 256 * 256, M); // M = T_out@Wo
        vec_kernel<<<1, 256, 0, stream>>>(Tout, wa_feat + l * 512,
                                          Wf + l * 256 * 256, bf + l * 256, tvec, u, cps);
        layer_kernel<<<NBLK, 256, 0, stream>>>(h, M, u, cps, tvec, ba_feat + l,
                                               bo + l * 256, lng + l * 256, lnb + l * 256);
    }

    bigmm_kernel<true><<<NBLK, 256, 0, stream>>>(h, W_out, b_out, ln_g, ln_b, h);
}